// PointNetSetAbstraction_31980326486607
// MI455X (gfx1250) — compile-verified
//
#include <hip/hip_runtime.h>

#define BB   16
#define NN   4096
#define SS   1024
#define KNB  32
#define FDIM 64
#define RAD2 0.04f
#define STRIDE 66   // padded LDS row stride (floats), even for b64 alignment

typedef __attribute__((ext_vector_type(2))) float v2f;
typedef __attribute__((ext_vector_type(8))) float v8f;

// ---------------------------------------------------------------------------
// Kernel 1: farthest point sampling. One block per batch, 1024 threads.
// Writes new_xyz workspace [B,S,3] and transposed new_xyz_out [B,3,S].
// ---------------------------------------------------------------------------
__global__ __launch_bounds__(1024) void fps_kernel(const float* __restrict__ xyz,
                                                   float* __restrict__ new_xyz_ws,
                                                   float* __restrict__ out_xyz) {
  const int b = blockIdx.x;
  const float* __restrict__ p = xyz + (size_t)b * NN * 3;

  __shared__ float min_d[NN];
  __shared__ float red_v[32];
  __shared__ int   red_i[32];
  __shared__ int   s_last;

  const int t = threadIdx.x;
  const int lane = t & 31, wave = t >> 5;

  for (int i = t; i < NN; i += 1024) min_d[i] = 1e10f;
  if (t == 0) s_last = 0;
  __syncthreads();

  for (int s = 0; s < SS; ++s) {
    const int last = s_last;
    const float cx = p[last * 3 + 0];
    const float cy = p[last * 3 + 1];
    const float cz = p[last * 3 + 2];
    if (t == 0) {
      new_xyz_ws[((size_t)b * SS + s) * 3 + 0] = cx;
      new_xyz_ws[((size_t)b * SS + s) * 3 + 1] = cy;
      new_xyz_ws[((size_t)b * SS + s) * 3 + 2] = cz;
      out_xyz[((size_t)b * 3 + 0) * SS + s] = cx;
      out_xyz[((size_t)b * 3 + 1) * SS + s] = cy;
      out_xyz[((size_t)b * 3 + 2) * SS + s] = cz;
    }
    float bestv = -1.0f;
    int   besti = 0;
    for (int i = t; i < NN; i += 1024) {
      const float dx = p[i * 3 + 0] - cx;
      const float dy = p[i * 3 + 1] - cy;
      const float dz = p[i * 3 + 2] - cz;
      const float d  = dx * dx + dy * dy + dz * dz;
      const float m  = fminf(min_d[i], d);
      min_d[i] = m;
      if (m > bestv) { bestv = m; besti = i; }  // strict > keeps lowest i per thread
    }
    // wave32 argmax reduce (prefer lower index on ties, matching jnp.argmax)
    #pragma unroll
    for (int off = 16; off > 0; off >>= 1) {
      const float ov = __shfl_down(bestv, off, 32);
      const int   oi = __shfl_down(besti, off, 32);
      if (ov > bestv || (ov == bestv && oi < besti)) { bestv = ov; besti = oi; }
    }
    if (lane == 0) { red_v[wave] = bestv; red_i[wave] = besti; }
    __syncthreads();
    if (t == 0) {
      float bv = red_v[0]; int bi = red_i[0];
      for (int w = 1; w < 32; ++w)
        if (red_v[w] > bv || (red_v[w] == bv && red_i[w] < bi)) { bv = red_v[w]; bi = red_i[w]; }
      s_last = bi;
    }
    __syncthreads();
  }
}

// ---------------------------------------------------------------------------
// Kernel 2: ball query + gather + MLP(64->64->128, relu) + max over K.
// 64 threads/block = 2 waves, one query per wave. No cross-wave barriers.
// ---------------------------------------------------------------------------
__global__ __launch_bounds__(64) void ballmlp_kernel(const float* __restrict__ xyz,
                                                     const float* __restrict__ features,
                                                     const float* __restrict__ W1,
                                                     const float* __restrict__ b1,
                                                     const float* __restrict__ W2,
                                                     const float* __restrict__ b2,
                                                     const float* __restrict__ new_xyz_ws,
                                                     float* __restrict__ out_pts) {
  __shared__ float featA[2][KNB * STRIDE];
  __shared__ float h1m[2][KNB * STRIDE];
  __shared__ int   nbr[2][KNB];

  const int wave = threadIdx.x >> 5;
  const int lane = threadIdx.x & 31;
  const int q = blockIdx.x * 2 + wave;       // global query, [0, B*S)
  const int b = q / SS;
  const int s = q % SS;

  // ---- ball query: lowest-index points within radius, early exit -----------
  const float qx = new_xyz_ws[(size_t)q * 3 + 0];
  const float qy = new_xyz_ws[(size_t)q * 3 + 1];
  const float qz = new_xyz_ws[(size_t)q * 3 + 2];
  const float* __restrict__ px = xyz + (size_t)b * NN * 3;

  if (lane == 0) nbr[wave][0] = 0;
  int count = 0;
  for (int base = 0; base < NN && count < KNB; base += 32) {
    const int i = base + lane;
    const float dx = px[i * 3 + 0] - qx;
    const float dy = px[i * 3 + 1] - qy;
    const float dz = px[i * 3 + 2] - qz;
    const float d2 = dx * dx + dy * dy + dz * dz;
    const bool hit = (d2 <= RAD2);
    const unsigned mask = (unsigned)__ballot(hit);   // wave32: low 32 bits
    if (hit) {
      const int slot = count + __popc(mask & ((1u << lane) - 1u));
      if (slot < KNB) nbr[wave][slot] = i;
    }
    count += __popc(mask);                            // uniform across wave
  }
  const int cnt = count < KNB ? count : KNB;
  const int myidx = (lane < cnt) ? nbr[wave][lane] : nbr[wave][0];  // pad w/ first

  // ---- gather features[b, myidx, :] -> LDS row `lane` ----------------------
  {
    const float* __restrict__ frow = features + ((size_t)b * NN + myidx) * FDIM;
    float* fA = &featA[wave][lane * STRIDE];
    #pragma unroll
    for (int d = 0; d < FDIM; d += 4) {
      const float4 v = *(const float4*)(frow + d);
      *(float2*)(fA + d)     = make_float2(v.x, v.y);
      *(float2*)(fA + d + 2) = make_float2(v.z, v.w);
    }
  }

  const int nlo   = lane & 15;          // column within 16-wide N tile
  const int khalf = (lane >> 4) << 1;   // lanes 16-31 supply K+2,K+3

  // ---- MLP1: h1[k,e] = relu(feat[k,:] . W1[e,:] + b1[e]), e in [0,64) ------
  for (int et = 0; et < 4; ++et) {
    const int col = et * 16 + nlo;
    const float bias = b1[col];
    const float* __restrict__ wbase = W1 + (size_t)col * FDIM + khalf;
    for (int mt = 0; mt < 2; ++mt) {
      const float* abase = &featA[wave][(mt * 16 + nlo) * STRIDE + khalf];
      v8f acc = {};
      #pragma unroll
      for (int kk = 0; kk < 16; ++kk) {
        const v2f a = *(const v2f*)(abase + kk * 4);
        const v2f w = *(const v2f*)(wbase + kk * 4);
        acc = __builtin_amdgcn_wmma_f32_16x16x4_f32(false, a, false, w,
                                                    (short)0, acc, false, false);
      }
      const int rbase = mt * 16 + ((lane >> 4) << 3);  // D: lanes>=16 hold rows M+8
      #pragma unroll
      for (int j = 0; j < 8; ++j) {
        float v = acc[j] + bias;
        v = v > 0.0f ? v : 0.0f;
        h1m[wave][(rbase + j) * STRIDE + col] = v;
      }
    }
  }

  // ---- MLP2 + max over K: out[e2] = max_k relu(h1[k,:] . W2[e2,:] + b2[e2]) -
  for (int et = 0; et < 8; ++et) {
    const int col = et * 16 + nlo;
    const float bias = b2[col];
    const float* __restrict__ wbase = W2 + (size_t)col * FDIM + khalf;
    float m = 0.0f;                      // relu output >= 0
    for (int mt = 0; mt < 2; ++mt) {
      const float* abase = &h1m[wave][(mt * 16 + nlo) * STRIDE + khalf];
      v8f acc = {};
      #pragma unroll
      for (int kk = 0; kk < 16; ++kk) {
        const v2f a = *(const v2f*)(abase + kk * 4);
        const v2f w = *(const v2f*)(wbase + kk * 4);
        acc = __builtin_amdgcn_wmma_f32_16x16x4_f32(false, a, false, w,
                                                    (short)0, acc, false, false);
      }
      #pragma unroll
      for (int j = 0; j < 8; ++j) {
        float v = acc[j] + bias;
        v = v > 0.0f ? v : 0.0f;
        m = fmaxf(m, v);
      }
    }
    // column `col`'s 32 rows live in lanes nlo and nlo+16: fold halves
    const float o = __shfl_xor(m, 16, 32);
    m = fmaxf(m, o);
    if (lane < 16) {
      out_pts[((size_t)b * 128 + col) * SS + s] = m;
    }
  }
}

// ---------------------------------------------------------------------------
extern "C" void kernel_launch(void* const* d_in, const int* in_sizes, int n_in,
                              void* d_out, int out_size, void* d_ws, size_t ws_size,
                              hipStream_t stream) {
  const float* xyz      = (const float*)d_in[0];   // [16,4096,3]
  const float* features = (const float*)d_in[1];   // [16,4096,64]
  const float* W1       = (const float*)d_in[2];   // [64,64]
  const float* b1       = (const float*)d_in[3];   // [64]
  const float* W2       = (const float*)d_in[4];   // [128,64]
  const float* b2       = (const float*)d_in[5];   // [128]

  float* out        = (float*)d_out;               // [B,3,S] then [B,128,S]
  float* new_xyz_ws = (float*)d_ws;                // [B,S,3] scratch

  fps_kernel<<<BB, 1024, 0, stream>>>(xyz, new_xyz_ws, out);

  float* out_pts = out + (size_t)BB * 3 * SS;      // offset 49152
  ballmlp_kernel<<<(BB * SS) / 2, 64, 0, stream>>>(xyz, features, W1, b1, W2, b2,
                                                   new_xyz_ws, out_pts);
}